// TVL1OF_10909216931977
// MI455X (gfx1250) — compile-verified
//
#include <hip/hip_runtime.h>
#include <cstdint>

constexpr int kB  = 4;
constexpr int kH  = 512;
constexpr int kW  = 512;
constexpr int kHW = kH * kW;
constexpr int kIter = 15;
constexpr int kGX = kW / 32;          // 16
constexpr int kGY = kH / 8;           // 64
constexpr int kPart = kGX * kGY * kB; // 4096 blocks
constexpr int kTP = 34;               // tile pitch (32 + 2 halo)
constexpr int kTR = 10;               // tile rows  (8 + 2 halo)
constexpr int kTel = kTP * kTR;       // 340 elems per channel

// ---------------------------------------------------------------------------
// CDNA5 async global->LDS staging of a (10 x 34) halo tile of both u channels.
// In-bounds lanes issue global_load_async_to_lds_b32 (tracked by ASYNCcnt);
// out-of-bounds lanes zero their LDS slot with a normal ds store (disjoint
// addresses). Then s_wait_asynccnt 0 + workgroup barrier.
// ---------------------------------------------------------------------------
__device__ __forceinline__ void async_tile_load(float* __restrict__ tile,
                                                const float* __restrict__ u,
                                                int b, int i0, int j0, int t) {
#pragma unroll
  for (int ch = 0; ch < 2; ++ch) {
    const float* up = u + (size_t)(b * 2 + ch) * kHW;
    float* tch = tile + ch * kTel;
#pragma unroll
    for (int base = 0; base < 512; base += 256) {
      int e = base + t;
      if (e < kTel) {
        int rr = e / kTP;
        int cc = e - rr * kTP;
        int gi = i0 - 1 + rr;
        int gj = j0 - 1 + cc;
        unsigned laddr = (unsigned)(uintptr_t)(const void*)(tch + e);
        if (gi >= 0 && gi < kH && gj >= 0 && gj < kW) {
          const float* gp = up + gi * kW + gj;
          asm volatile("global_load_async_to_lds_b32 %0, %1, off"
                       :: "v"(laddr), "v"(gp) : "memory");
        } else {
          tch[e] = 0.0f;  // zero-pad halo
        }
      }
    }
  }
  asm volatile("s_wait_asynccnt 0x0" ::: "memory");
  __syncthreads();
}

// Sobel (cross-correlation with GX/GY) at the tile position whose (-1,-1)
// corner is tc. GX=[[1,0,-1],[2,0,-2],[1,0,-1]], GY=[[1,2,1],[0,0,0],[-1,-2,-1]]
__device__ __forceinline__ void sobel_at(const float* __restrict__ tc,
                                         float& gx, float& gy) {
  float m00 = tc[0],        m01 = tc[1],        m02 = tc[2];
  float m10 = tc[kTP],                          m12 = tc[kTP + 2];
  float m20 = tc[2 * kTP],  m21 = tc[2 * kTP+1],m22 = tc[2 * kTP + 2];
  gx = (m00 - m02) + 2.0f * (m10 - m12) + (m20 - m22);
  gy = (m00 + 2.0f * m01 + m02) - (m20 + 2.0f * m21 + m22);
}

// ---------------------------------------------------------------------------
// Init: zero p1, p2, u (u lives in d_out; harness poisons buffers with 0xAA)
// ---------------------------------------------------------------------------
__global__ void k_init(float* __restrict__ p1, float* __restrict__ p2,
                       float* __restrict__ u, int n) {
  int i = blockIdx.x * blockDim.x + threadIdx.x;
  if (i < n) { p1[i] = 0.0f; p2[i] = 0.0f; u[i] = 0.0f; }
}

// ---------------------------------------------------------------------------
// Kernel A: pointwise u update. Recomputes image Sobel / norm_grad / rho_c
// from x (loop-invariant, L2-resident) to keep workspace small.
// ---------------------------------------------------------------------------
__global__ __launch_bounds__(256) void k_update_u(
    const float* __restrict__ x, float* __restrict__ u,
    const float* __restrict__ p1, const float* __restrict__ p2,
    const float* __restrict__ wx, const float* __restrict__ wy,
    const float* __restrict__ lamp, const float* __restrict__ thetap) {
  const int j = blockIdx.x * 32 + threadIdx.x;
  const int i = blockIdx.y * 8 + threadIdx.y;
  const int b = blockIdx.z;
  const float lam = lamp[0], theta = thetap[0];
  const float wx0 = wx[0], wx1 = wx[1], wy0 = wy[0], wy1 = wy[1];

  const float* x0p = x + (size_t)(b * 2 + 0) * kHW;
  const float* x1p = x + (size_t)(b * 2 + 1) * kHW;

  float t[3][3];
#pragma unroll
  for (int dy = 0; dy < 3; ++dy)
#pragma unroll
    for (int dx = 0; dx < 3; ++dx) {
      int ii = i + dy - 1, jj = j + dx - 1;
      t[dy][dx] = (ii >= 0 && ii < kH && jj >= 0 && jj < kW)
                      ? x1p[ii * kW + jj] : 0.0f;
    }
  float gx = (t[0][0] - t[0][2]) + 2.0f * (t[1][0] - t[1][2]) + (t[2][0] - t[2][2]);
  float gy = (t[0][0] + 2.0f * t[0][1] + t[0][2]) - (t[2][0] + 2.0f * t[2][1] + t[2][2]);
  float ng = gx * gx + gy * gy;

  const size_t idx = (size_t)i * kW + j;
  float* u0p = u + (size_t)(b * 2 + 0) * kHW;
  float* u1p = u + (size_t)(b * 2 + 1) * kHW;
  float u0 = u0p[idx], u1 = u1p[idx];

  float rho = (t[1][1] - x0p[idx]) + gx * u0 + gy * u1;
  float th  = theta * lam * ng;
  float ar  = fabsf(rho);
  float v0 = u0, v1 = u1;
  if (ar < th) {
    float sc = rho / ng;
    v0 -= sc * gx; v1 -= sc * gy;
  } else if (ar > th) {
    float s = (rho > 0.0f) ? 1.0f : ((rho < 0.0f) ? -1.0f : 0.0f);
    float tl = theta * lam * s;
    v0 -= tl * gx; v1 -= tl * gy;
  }

  const float* p1b = p1 + (size_t)b * 2 * kHW;
  const float* p2b = p2 + (size_t)b * 2 * kHW;
  // div: wx0*p1[j-1] + wx1*p1[j] + wy0*p2[i-1] + wy1*p2[i], zero-padded
  float d0 = wx1 * p1b[idx] + wy1 * p2b[idx];
  float d1 = wx1 * p1b[kHW + idx] + wy1 * p2b[kHW + idx];
  if (j > 0) { d0 += wx0 * p1b[idx - 1];       d1 += wx0 * p1b[kHW + idx - 1]; }
  if (i > 0) { d0 += wy0 * p2b[idx - kW];      d1 += wy0 * p2b[kHW + idx - kW]; }

  u0p[idx] = v0 + theta * d0;
  u1p[idx] = v1 + theta * d1;
}

// ---------------------------------------------------------------------------
// Kernel B: Sobel of both u channels (async-LDS tile), per-block |.| sums.
// ---------------------------------------------------------------------------
__global__ __launch_bounds__(256) void k_grad_reduce(
    const float* __restrict__ u, float2* __restrict__ part) {
  __shared__ float tile[2 * kTel];
  __shared__ float r0[256], r1[256];
  const int tx = threadIdx.x, ty = threadIdx.y;
  const int t = ty * 32 + tx;
  const int j0 = blockIdx.x * 32, i0 = blockIdx.y * 8, b = blockIdx.z;

  async_tile_load(tile, u, b, i0, j0, t);

  float gx0, gy0, gx1, gy1;
  sobel_at(tile + 0 * kTel + ty * kTP + tx, gx0, gy0);
  sobel_at(tile + 1 * kTel + ty * kTP + tx, gx1, gy1);

  r0[t] = fabsf(gx0) + fabsf(gy0);
  r1[t] = fabsf(gx1) + fabsf(gy1);
  __syncthreads();
#pragma unroll
  for (int s = 128; s > 0; s >>= 1) {
    if (t < s) { r0[t] += r0[t + s]; r1[t] += r1[t + s]; }
    __syncthreads();
  }
  if (t == 0) {
    int blk = (blockIdx.z * gridDim.y + blockIdx.y) * gridDim.x + blockIdx.x;
    part[blk] = make_float2(r0[0], r1[0]);
  }
}

// ---------------------------------------------------------------------------
// Kernel R: deterministic final reduction -> 1/(1+r*S1), 1/(1+r*S2)
// ---------------------------------------------------------------------------
__global__ __launch_bounds__(256) void k_finish(
    const float2* __restrict__ part, float* __restrict__ sinv,
    const float* __restrict__ taup, const float* __restrict__ thetap) {
  __shared__ float r0[256], r1[256];
  const int t = threadIdx.x;
  float a0 = 0.0f, a1 = 0.0f;
  for (int k = t; k < kPart; k += 256) { a0 += part[k].x; a1 += part[k].y; }
  r0[t] = a0; r1[t] = a1;
  __syncthreads();
#pragma unroll
  for (int s = 128; s > 0; s >>= 1) {
    if (t < s) { r0[t] += r0[t + s]; r1[t] += r1[t + s]; }
    __syncthreads();
  }
  if (t == 0) {
    float r = taup[0] / thetap[0];
    sinv[0] = 1.0f / (1.0f + r * r0[0]);
    sinv[1] = 1.0f / (1.0f + r * r1[0]);
  }
}

// ---------------------------------------------------------------------------
// Kernel C: in-place p1/p2 update (async-LDS tiled Sobel of u).
// p1[b,0]+=r*Gx(u0), p1[b,1]+=r*Gy(u0) scaled by inv1; p2 with u1, inv2.
// ---------------------------------------------------------------------------
__global__ __launch_bounds__(256) void k_update_p(
    float* __restrict__ p1, float* __restrict__ p2,
    const float* __restrict__ u, const float* __restrict__ sinv,
    const float* __restrict__ taup, const float* __restrict__ thetap) {
  __shared__ float tile[2 * kTel];
  const int tx = threadIdx.x, ty = threadIdx.y;
  const int t = ty * 32 + tx;
  const int j0 = blockIdx.x * 32, i0 = blockIdx.y * 8, b = blockIdx.z;

  async_tile_load(tile, u, b, i0, j0, t);

  const float r = taup[0] / thetap[0];
  const float inv1 = sinv[0], inv2 = sinv[1];
  float gx0, gy0, gx1, gy1;
  sobel_at(tile + 0 * kTel + ty * kTP + tx, gx0, gy0);
  sobel_at(tile + 1 * kTel + ty * kTP + tx, gx1, gy1);

  const size_t idx = (size_t)(i0 + ty) * kW + (j0 + tx);
  float* p1b = p1 + (size_t)b * 2 * kHW;
  float* p2b = p2 + (size_t)b * 2 * kHW;
  p1b[idx]       = (p1b[idx]       + r * gx0) * inv1;
  p1b[kHW + idx] = (p1b[kHW + idx] + r * gy0) * inv1;
  p2b[idx]       = (p2b[idx]       + r * gx1) * inv2;
  p2b[kHW + idx] = (p2b[kHW + idx] + r * gy1) * inv2;
}

// ---------------------------------------------------------------------------
extern "C" void kernel_launch(void* const* d_in, const int* in_sizes, int n_in,
                              void* d_out, int out_size, void* d_ws, size_t ws_size,
                              hipStream_t stream) {
  (void)in_sizes; (void)n_in; (void)out_size; (void)ws_size;
  const float* x     = (const float*)d_in[0];
  const float* wx    = (const float*)d_in[1];
  const float* wy    = (const float*)d_in[2];
  const float* lam   = (const float*)d_in[3];
  const float* tau   = (const float*)d_in[4];
  const float* theta = (const float*)d_in[5];
  float* u = (float*)d_out;  // u lives directly in d_out

  // workspace carve: p1 (8MB) | p2 (8MB) | partials (32KB) | sinv (8B)
  float*  p1   = (float*)d_ws;
  float*  p2   = p1 + (size_t)2 * kB * kHW;
  float2* part = (float2*)(p2 + (size_t)2 * kB * kHW);
  float*  sinv = (float*)(part + kPart);

  const int n = 2 * kB * kHW;
  k_init<<<(n + 255) / 256, 256, 0, stream>>>(p1, p2, u, n);

  dim3 blk(32, 8, 1), grd(kGX, kGY, kB);
  for (int it = 0; it < kIter; ++it) {
    k_update_u<<<grd, blk, 0, stream>>>(x, u, p1, p2, wx, wy, lam, theta);
    if (it + 1 < kIter) {  // p update feeds only the next iteration
      k_grad_reduce<<<grd, blk, 0, stream>>>(u, part);
      k_finish<<<1, 256, 0, stream>>>(part, sinv, tau, theta);
      k_update_p<<<grd, blk, 0, stream>>>(p1, p2, u, sinv, tau, theta);
    }
  }
}